// SGGM_6055903887543
// MI455X (gfx1250) — compile-verified
//
#include <hip/hip_runtime.h>

typedef __attribute__((ext_vector_type(2))) float v2f;
typedef __attribute__((ext_vector_type(8))) float v8f;

#define NB    32
#define NN    128
#define NH    128
#define EPER  4096
#define MTOT  (NB * EPER)      // 131072 edges
#define KTOT  384              // 3*H
#define NOUT  768              // 6*H
#define NSTRIPS 6              // 768 / 128 cols per wave
#define WAVES_PER_BLOCK 8

__global__ __launch_bounds__(256) void
SGGM_gather_gemm_wmma(const float* __restrict__ h_node,
                      const float* __restrict__ h_edge,
                      const int*   __restrict__ pairlist,
                      const float* __restrict__ W,
                      const float* __restrict__ bias,
                      float*       __restrict__ out) {
    const int tid  = threadIdx.x;
    const int lane = tid & 31;
    const int warp = tid >> 5;
    const int gw   = blockIdx.x * WAVES_PER_BLOCK + warp; // global wave id
    const int m2   = gw / NSTRIPS;   // 32-edge tile index (0..4095)
    const int ns   = gw % NSTRIPS;   // 128-col strip     (0..5)

    const int rlane   = lane & 15;              // M-row within 16-tile / N-col within 16-tile
    const int koff    = (lane < 16) ? 0 : 2;    // A/B K-pair select per half-wave (ISA 7.12.2)
    const int colbase = ns * 128;

    // ---- per-lane gather bases for the two 16-edge subtiles ----
    const int r0 = m2 * 32 + rlane;
    const int r1 = r0 + 16;

    const int b0 = pairlist[3 * r0 + 0];
    const int i0 = pairlist[3 * r0 + 1];
    const int j0 = pairlist[3 * r0 + 2];
    const int b1 = pairlist[3 * r1 + 0];
    const int i1 = pairlist[3 * r1 + 1];
    const int j1 = pairlist[3 * r1 + 2];

    const float* pNi0 = h_node + ((size_t)b0 * NN + i0) * NH;
    const float* pNj0 = h_node + ((size_t)b0 * NN + j0) * NH;
    const float* pE0  = h_edge + (((size_t)b0 * NN + i0) * NN + j0) * NH;
    const float* pNi1 = h_node + ((size_t)b1 * NN + i1) * NH;
    const float* pNj1 = h_node + ((size_t)b1 * NN + j1) * NH;
    const float* pE1  = h_edge + (((size_t)b1 * NN + i1) * NN + j1) * NH;

    // ---- init accumulators with bias (C/D layout: N = lane%16, M = vgpr + 8*(lane>=16)) ----
    v8f acc[16];
#pragma unroll
    for (int t = 0; t < 8; ++t) {
        const float c = bias[colbase + t * 16 + rlane];
        v8f cv = {c, c, c, c, c, c, c, c};
        acc[t]     = cv;
        acc[t + 8] = cv;
    }

    // ---- K loop: 3 gather segments x 128, step 4 ----
#pragma unroll
    for (int seg = 0; seg < 3; ++seg) {
        const float* base0 = (seg == 0) ? pNi0 : (seg == 1) ? pNj0 : pE0;
        const float* base1 = (seg == 0) ? pNi1 : (seg == 1) ? pNj1 : pE1;
        const int    kbase = seg * NH;
        for (int k = 0; k < NH; k += 4) {
            // A 16x4 f32: lane L holds A[L%16][k+koff], A[L%16][k+koff+1]
            const v2f a0 = *(const v2f*)(base0 + k + koff);
            const v2f a1 = *(const v2f*)(base1 + k + koff);
            // B 4x16 f32: lane L holds W[k+koff][col], W[k+koff+1][col], col = colbase + t*16 + L%16
            const float* wp = W + (size_t)(kbase + k + koff) * NOUT + colbase + rlane;
#pragma unroll
            for (int t = 0; t < 8; ++t) {
                v2f bv = { wp[t * 16], wp[NOUT + t * 16] };
                acc[t] = __builtin_amdgcn_wmma_f32_16x16x4_f32(
                    false, a0, false, bv, (short)0, acc[t], false, false);
                acc[t + 8] = __builtin_amdgcn_wmma_f32_16x16x4_f32(
                    false, a1, false, bv, (short)0, acc[t + 8], false, false);
            }
        }
    }

    // ---- store D (non-temporal: 403 MB streaming output, keep L2 for W/h_node) ----
    const int moff = (lane < 16) ? 0 : 8;
#pragma unroll
    for (int t = 0; t < 8; ++t) {
        const int ocol = colbase + t * 16 + rlane;
#pragma unroll
        for (int v = 0; v < 8; ++v) {
            const size_t orow0 = (size_t)(m2 * 32 + v + moff);
            __builtin_nontemporal_store(acc[t][v],     out + orow0 * NOUT + ocol);
            __builtin_nontemporal_store(acc[t + 8][v], out + (orow0 + 16) * NOUT + ocol);
        }
    }
}

extern "C" void kernel_launch(void* const* d_in, const int* in_sizes, int n_in,
                              void* d_out, int out_size, void* d_ws, size_t ws_size,
                              hipStream_t stream) {
    const float* h_node   = (const float*)d_in[0];
    const float* h_edge   = (const float*)d_in[1];
    const int*   pairlist = (const int*)d_in[2];
    const float* W        = (const float*)d_in[3];
    const float* bias     = (const float*)d_in[4];
    float*       out      = (float*)d_out;

    const int total_waves = (MTOT / 32) * NSTRIPS;         // 24576
    const int blocks      = total_waves / WAVES_PER_BLOCK; // 3072
    SGGM_gather_gemm_wmma<<<blocks, 256, 0, stream>>>(h_node, h_edge, pairlist, W, bias, out);
}